// DecoderRNN_34419867910480
// MI455X (gfx1250) — compile-verified
//
#include <hip/hip_runtime.h>
#include <hip/hip_bf16.h>
#include <stdint.h>

#define Vv 50000
#define Ec 256
#define Hc 256
#define Bc 64
#define Sc 2048
#define TS 64            // s-tile rows per workgroup in attention GEMM
#define LDP 260          // padded LDS row stride (floats): 260 % 64 = 4 -> conflict-free

typedef __attribute__((ext_vector_type(2))) float v2f;
typedef __attribute__((ext_vector_type(8))) float v8f;

// Low 32 bits of a flat pointer to __shared__ = LDS byte offset (LDS aperture
// truncates addr[31:0]); that is what GLOBAL_LOAD_ASYNC_TO_LDS wants in VDST.
#define LDS_OFF(p) ((uint32_t)(uintptr_t)(p))

// Async copy 16B global -> LDS, tracked by ASYNCcnt (CDNA5 path).
__device__ __forceinline__ void async_b128(uint32_t lds_byte_off, const void* gsrc) {
    asm volatile("global_load_async_to_lds_b128 %0, %1, off"
                 :: "v"(lds_byte_off), "v"(gsrc) : "memory");
}
__device__ __forceinline__ void wait_async0() {
    asm volatile("s_wait_asynccnt 0x0" ::: "memory");
}

// ---------------------------------------------------------------------------
// Kernel 0: gather word embeddings  x[b][e] = emb[inp[b]][e]
// ---------------------------------------------------------------------------
__global__ void gather_emb_kernel(const int64_t* __restrict__ inp,
                                  const float* __restrict__ emb,
                                  float* __restrict__ x) {
    int b = blockIdx.x;
    int e = threadIdx.x;
    int64_t row = inp[b];
    x[b * Ec + e] = emb[row * Ec + e];
}

// ---------------------------------------------------------------------------
// Kernel 1: wu[b][h] = sum_e x[b][e] * U[e][h]
// ---------------------------------------------------------------------------
__global__ __launch_bounds__(256)
void wu_kernel(const float* __restrict__ x, const float* __restrict__ U,
               float* __restrict__ wu) {
    __shared__ float xs[Ec];
    int b = blockIdx.x;
    int h = threadIdx.x;
    xs[h] = x[b * Ec + h];
    __syncthreads();
    float acc = 0.f;
    #pragma unroll 8
    for (int e = 0; e < Ec; ++e)
        acc += xs[e] * U[e * Hc + h];
    wu[b * Hc + h] = acc;
}

// ---------------------------------------------------------------------------
// Kernel 2: attention scores (17 GFLOP fp32 WMMA GEMM, fused tanh + dot-v)
//   e[b][s] = sum_h tanh(wu[b][h] + sum_k enc[b][s][k]*W[k][h]) * v[h]
// enc tile (64x256) + W k-panels staged via global_load_async_to_lds_b128.
// ---------------------------------------------------------------------------
__global__ __launch_bounds__(256)
void attn_scores_kernel(const float* __restrict__ enc, const float* __restrict__ W,
                        const float* __restrict__ wu, const float* __restrict__ v,
                        float* __restrict__ e_out) {
    __shared__ float lds_enc[TS * LDP];   // 66,560 B
    __shared__ float lds_w[64 * LDP];     // 66,560 B
    __shared__ float lds_wu[Hc];
    __shared__ float lds_v[Hc];
    __shared__ float e_red[TS];

    int b   = blockIdx.x;
    int s0  = blockIdx.y * TS;
    int tid = threadIdx.x;
    int lane = tid & 31;
    int wave = tid >> 5;
    int l15  = lane & 15;
    int khalf = lane >> 4;

    if (tid < Hc) { lds_wu[tid] = wu[b * Hc + tid]; lds_v[tid] = v[tid]; }
    if (tid < TS) e_red[tid] = 0.f;

    // async-stage enc tile: 64 rows x 256 floats (16B chunks, padded rows)
    {
        const float* encb = enc + ((size_t)b * Sc + s0) * Hc;
        for (int i = tid; i < TS * 64; i += 256) {
            int r = i >> 6, c = i & 63;
            async_b128(LDS_OFF(&lds_enc[r * LDP + c * 4]), encb + r * Hc + c * 4);
        }
    }

    int mtile = wave & 3;        // rows mtile*16 .. +15
    int nbase = (wave >> 2) * 128;

    v8f acc[8];
    #pragma unroll
    for (int t = 0; t < 8; ++t) acc[t] = (v8f){0,0,0,0,0,0,0,0};

    for (int kb = 0; kb < Hc; kb += 64) {
        __syncthreads();                 // lds_w free (prev compute done)
        // async-stage W k-panel: rows kb..kb+63, 256 cols
        const float* wb = W + (size_t)kb * Hc;
        for (int i = tid; i < 64 * 64; i += 256) {
            int r = i >> 6, c = i & 63;
            async_b128(LDS_OFF(&lds_w[r * LDP + c * 4]), wb + r * Hc + c * 4);
        }
        wait_async0();                   // enc (first iter) + this panel landed
        __syncthreads();

        #pragma unroll 4
        for (int k = 0; k < 64; k += 4) {
            const float* ap = &lds_enc[(mtile * 16 + l15) * LDP + kb + k + 2 * khalf];
            v2f a; a.x = ap[0]; a.y = ap[1];
            #pragma unroll
            for (int t = 0; t < 8; ++t) {
                const float* bp = &lds_w[(k + 2 * khalf) * LDP + nbase + t * 16 + l15];
                v2f bf; bf.x = bp[0]; bf.y = bp[LDP];
                acc[t] = __builtin_amdgcn_wmma_f32_16x16x4_f32(
                    false, a, false, bf, (short)0, acc[t], false, false);
            }
        }
    }

    // epilogue: tanh(wu + acc) * v, reduce over columns
    float partial[8];
    #pragma unroll
    for (int r = 0; r < 8; ++r) partial[r] = 0.f;
    #pragma unroll
    for (int t = 0; t < 8; ++t) {
        int ncol = nbase + t * 16 + l15;
        float wuv = lds_wu[ncol];
        float vv  = lds_v[ncol];
        #pragma unroll
        for (int r = 0; r < 8; ++r)
            partial[r] += tanhf(acc[t][r] + wuv) * vv;
    }
    // butterfly over 16 lanes of each half (rows 0-7 in lanes 0-15, 8-15 in 16-31)
    #pragma unroll
    for (int m = 8; m >= 1; m >>= 1)
        #pragma unroll
        for (int r = 0; r < 8; ++r)
            partial[r] += __shfl_xor(partial[r], m, 32);

    if (l15 == 0) {
        int rowbase = mtile * 16 + 8 * khalf;
        #pragma unroll
        for (int r = 0; r < 8; ++r)
            atomicAdd(&e_red[rowbase + r], partial[r]);   // ds_add_f32
    }
    __syncthreads();
    if (tid < TS) e_out[(size_t)b * Sc + s0 + tid] = e_red[tid];
}

// ---------------------------------------------------------------------------
// Kernel 3a: softmax over S -> alpha
// ---------------------------------------------------------------------------
__global__ __launch_bounds__(256)
void softmax_kernel(const float* __restrict__ e, float* __restrict__ alpha) {
    __shared__ float red[256];
    int b = blockIdx.x;
    int tid = threadIdx.x;

    float m = -1e30f;
    for (int s = tid; s < Sc; s += 256) m = fmaxf(m, e[(size_t)b * Sc + s]);
    red[tid] = m; __syncthreads();
    for (int off = 128; off; off >>= 1) {
        if (tid < off) red[tid] = fmaxf(red[tid], red[tid + off]);
        __syncthreads();
    }
    float mx = red[0]; __syncthreads();

    float sum = 0.f;
    for (int s = tid; s < Sc; s += 256) sum += expf(e[(size_t)b * Sc + s] - mx);
    red[tid] = sum; __syncthreads();
    for (int off = 128; off; off >>= 1) {
        if (tid < off) red[tid] += red[tid + off];
        __syncthreads();
    }
    float inv = 1.f / red[0];
    for (int s = tid; s < Sc; s += 256)
        alpha[(size_t)b * Sc + s] = expf(e[(size_t)b * Sc + s] - mx) * inv;
}

// ---------------------------------------------------------------------------
// Kernel 3b: partial context over an s-chunk of 256 (grid (B, 8) -> 512 blocks
// streaming enc from L2)
// ---------------------------------------------------------------------------
__global__ __launch_bounds__(256)
void context_partial_kernel(const float* __restrict__ alpha, const float* __restrict__ enc,
                            float* __restrict__ cpart) {
    __shared__ float al[256];
    int b = blockIdx.x, chunk = blockIdx.y, tid = threadIdx.x;
    int sbase = chunk * 256;
    al[tid] = alpha[(size_t)b * Sc + sbase + tid];
    __syncthreads();
    float c = 0.f;
    const float* ebase = enc + ((size_t)b * Sc + sbase) * Hc + tid;
    #pragma unroll 8
    for (int s = 0; s < 256; ++s)
        c += al[s] * ebase[(size_t)s * Hc];
    cpart[((size_t)b * 8 + chunk) * Hc + tid] = c;
}

// ---------------------------------------------------------------------------
// Kernel 4: h_new = tanh(x@W_ih^T + b_ih + context@W_hh^T + b_hh)
// (context = sum of 8 partials, folded into staging)
// ---------------------------------------------------------------------------
__global__ __launch_bounds__(256)
void hnew_kernel(const float* __restrict__ x, const float* __restrict__ cpart,
                 const float* __restrict__ W_ih, const float* __restrict__ W_hh,
                 const float* __restrict__ b_ih, const float* __restrict__ b_hh,
                 float* __restrict__ hnew_ws, float* __restrict__ hnew_out) {
    __shared__ float xs[Ec];
    __shared__ float cs[Hc];
    int b = blockIdx.x;
    int h = threadIdx.x;
    xs[h] = x[b * Ec + h];
    float c = 0.f;
    #pragma unroll
    for (int j = 0; j < 8; ++j)
        c += cpart[((size_t)b * 8 + j) * Hc + h];
    cs[h] = c;
    __syncthreads();
    float a = b_ih[h] + b_hh[h];
    #pragma unroll 8
    for (int k = 0; k < Hc; ++k)
        a += xs[k] * W_ih[h * Ec + k] + cs[k] * W_hh[h * Hc + k];
    float t = tanhf(a);
    hnew_ws[b * Hc + h] = t;
    hnew_out[b * Hc + h] = t;
}

// ---------------------------------------------------------------------------
// Kernel 5: logits = h_new @ W_out^T + b_out via WMMA; async LDS staging for
// full tiles, guarded scalar staging for the one partial tail tile.
// ---------------------------------------------------------------------------
__global__ __launch_bounds__(256)
void logits_kernel(const float* __restrict__ hnew, const float* __restrict__ W_out,
                   const float* __restrict__ b_out, float* __restrict__ logits) {
    __shared__ float lds_h[Bc * LDP];   // h_new 64x256
    __shared__ float lds_w[64 * LDP];   // W_out rows n0..n0+63 x 256
    int n0  = blockIdx.x * 64;
    int tid = threadIdx.x;
    int lane = tid & 31;
    int wave = tid >> 5;
    int l15  = lane & 15;
    int khalf = lane >> 4;
    bool full = (n0 + 64 <= Vv);        // uniform per block

    // stage h_new (async)
    for (int i = tid; i < Bc * 64; i += 256) {
        int r = i >> 6, c = i & 63;
        async_b128(LDS_OFF(&lds_h[r * LDP + c * 4]), hnew + r * Hc + c * 4);
    }
    // stage W_out rows n0..n0+63
    if (full) {
        const float* wb = W_out + (size_t)n0 * Hc;
        for (int i = tid; i < 64 * 64; i += 256) {
            int r = i >> 6, c = i & 63;
            async_b128(LDS_OFF(&lds_w[r * LDP + c * 4]), wb + (size_t)r * Hc + c * 4);
        }
    } else {
        const float4* wv = (const float4*)W_out;
        for (int i = tid; i < 64 * 64; i += 256) {
            int r = i >> 6, c = i & 63;
            int n = n0 + r;
            float4 d = (n < Vv) ? wv[(size_t)n * 64 + c] : make_float4(0.f, 0.f, 0.f, 0.f);
            float* dst = &lds_w[r * LDP + c * 4];
            dst[0] = d.x; dst[1] = d.y; dst[2] = d.z; dst[3] = d.w;
        }
    }
    wait_async0();
    __syncthreads();

    int mtile = wave & 3;          // b rows mtile*16..
    int nbase = (wave >> 2) * 32;  // two 16-wide n-tiles per wave

    v8f acc[2];
    acc[0] = (v8f){0,0,0,0,0,0,0,0};
    acc[1] = (v8f){0,0,0,0,0,0,0,0};

    #pragma unroll 4
    for (int k = 0; k < Hc; k += 4) {
        const float* ap = &lds_h[(mtile * 16 + l15) * LDP + k + 2 * khalf];
        v2f a; a.x = ap[0]; a.y = ap[1];
        #pragma unroll
        for (int t = 0; t < 2; ++t) {
            // B[k][n] = W_out[n0+n][k] -> transposed read from padded LDS
            const float* bp = &lds_w[(nbase + t * 16 + l15) * LDP + k + 2 * khalf];
            v2f bf; bf.x = bp[0]; bf.y = bp[1];
            acc[t] = __builtin_amdgcn_wmma_f32_16x16x4_f32(
                false, a, false, bf, (short)0, acc[t], false, false);
        }
    }

    #pragma unroll
    for (int t = 0; t < 2; ++t) {
        int n = n0 + nbase + t * 16 + l15;
        if (n < Vv) {
            float bias = b_out[n];
            #pragma unroll
            for (int r = 0; r < 8; ++r) {
                int brow = mtile * 16 + r + 8 * khalf;
                logits[(size_t)brow * Vv + n] = acc[t][r] + bias;
            }
        }
    }
}

// ---------------------------------------------------------------------------
extern "C" void kernel_launch(void* const* d_in, const int* in_sizes, int n_in,
                              void* d_out, int out_size, void* d_ws, size_t ws_size,
                              hipStream_t stream) {
    const int64_t* inp  = (const int64_t*)d_in[0];
    // d_in[1] = hidden : unused by the reference computation
    const float* enc    = (const float*)d_in[2];
    const float* emb    = (const float*)d_in[3];
    const float* U      = (const float*)d_in[4];
    const float* W      = (const float*)d_in[5];
    const float* v      = (const float*)d_in[6];
    const float* W_ih   = (const float*)d_in[7];
    const float* W_hh   = (const float*)d_in[8];
    const float* b_ih   = (const float*)d_in[9];
    const float* b_hh   = (const float*)d_in[10];
    const float* W_out  = (const float*)d_in[11];
    const float* b_out  = (const float*)d_in[12];

    float* logits   = (float*)d_out;                    // B*V
    float* hnew_out = (float*)d_out + (size_t)Bc * Vv;  // + B*H

    // workspace layout (floats)
    float* ws      = (float*)d_ws;
    float* x       = ws;                 // 16384
    float* wu      = x + Bc * Ec;        // 16384
    float* e_sc    = wu + Bc * Hc;       // 131072
    float* alpha   = e_sc + Bc * Sc;     // 131072
    float* cpart   = alpha + Bc * Sc;    // 131072 (B x 8 x H)
    float* hnew    = cpart + Bc * 8 * Hc;// 16384

    gather_emb_kernel<<<Bc, Ec, 0, stream>>>(inp, emb, x);
    wu_kernel<<<Bc, Hc, 0, stream>>>(x, U, wu);
    attn_scores_kernel<<<dim3(Bc, Sc / TS), 256, 0, stream>>>(enc, W, wu, v, e_sc);
    softmax_kernel<<<Bc, 256, 0, stream>>>(e_sc, alpha);
    context_partial_kernel<<<dim3(Bc, 8), 256, 0, stream>>>(alpha, enc, cpart);
    hnew_kernel<<<Bc, Hc, 0, stream>>>(x, cpart, W_ih, W_hh, b_ih, b_hh, hnew, hnew_out);
    logits_kernel<<<(Vv + 63) / 64, 256, 0, stream>>>(hnew, W_out, b_out, logits);
}